// T_aT_R1_GCN_SSL_82695300317788
// MI455X (gfx1250) — compile-verified
//
#include <hip/hip_runtime.h>
#include <stdint.h>

// ---------------------------------------------------------------------------
// Problem constants (match reference)
// ---------------------------------------------------------------------------
#define D_DIM   128
#define FAN     10
#define N_B     1024
#define N_N1    (N_B * FAN)     // 10240
#define N_N0    (N_N1 * FAN)    // 102400
#define N_RS    512
#define N_R0    (N_RS * FAN)    // 5120

#define TILE_M       16                    // dst rows per workgroup (one wave)
#define EDGES_TILE   (TILE_M * FAN)        // 160 edges per dst tile
#define STAGE_K      16                    // edges staged per stage
#define NSTAGES      (EDGES_TILE / STAGE_K)
#define LDS_PAD      4                     // dword pad to rotate LDS banks
#define ROW_F        (D_DIM + LDS_PAD)     // 132 floats per LDS row
#define ROW_BYTES    (ROW_F * 4)           // 528 (16B-aligned)

typedef __attribute__((ext_vector_type(2))) float v2f;
typedef __attribute__((ext_vector_type(8))) float v8f;

// ---------------------------------------------------------------------------
// CDNA5 async Global->LDS copy, GVS form: mem = SGPR64 base + per-lane voff.
// VDST = per-lane LDS byte address.  Tracked with ASYNCcnt (in-order done).
// One call moves one 512B row (32 lanes x 16B).
// ---------------------------------------------------------------------------
__device__ __forceinline__ void async_row_b128(unsigned lds_addr, unsigned voff,
                                               unsigned long long sbase) {
    asm volatile("global_load_async_to_lds_b128 %0, %1, %2"
                 :: "v"(lds_addr), "v"(voff), "s"(sbase)
                 : "memory");
}

// ---------------------------------------------------------------------------
// Out-degree counting (src arrays are permutations here -> all-ones, but we
// compute it to preserve reference semantics).  Int atomics -> deterministic.
// ---------------------------------------------------------------------------
__global__ void count_deg_kernel(const int* __restrict__ src,
                                 int* __restrict__ deg, int n) {
    int i = blockIdx.x * blockDim.x + threadIdx.x;
    if (i < n) atomicAdd(&deg[src[i]], 1);
}

// ---------------------------------------------------------------------------
// Fused block_compute for 16 consecutive dst rows (edges contiguous per dst,
// dst = e / FAN).  Double-buffered async Global->LDS staging of raw src and
// rel rows; reduction via V_WMMA_F32_16X16X4_F32 with a selection A-matrix
// carrying the per-edge norm (rsqrt(out_deg)/sqrt(FAN)); the relation gating
// multiply happens at B-pack time.  Residual rows seed the C accumulator.
//
// f32 WMMA VGPR layouts (ISA 7.12.2), wave32:
//   A 16x4 : lane l -> M=l&15 ; vgpr0 = A[M][2*(l>>4)],  vgpr1 = A[M][1+2*(l>>4)]
//   B 4x16 : lane l -> N=l&15 ; vgpr0 = B[2*(l>>4)][N],  vgpr1 = B[1+2*(l>>4)][N]
//   C/D    : lane l -> N=l&15 ; vgpr j = C[j + 8*(l>>4)][N]
// ---------------------------------------------------------------------------
template <bool HAS_MAP, bool HAS_REL, bool HAS_RES>
__global__ __launch_bounds__(32)
void block_wmma_kernel(const float* __restrict__ src_rows,   // source row table
                       const int*   __restrict__ src_map,    // row = src_map[src_perm[e]]
                       const int*   __restrict__ src_perm,   // edge -> sampled src id
                       const float* __restrict__ rel_table,  // relation embeddings
                       const int*   __restrict__ rel_idx,    // edge -> relation id
                       const int*   __restrict__ deg,        // out-degree per sampled src id
                       const float* __restrict__ res_rows,   // residual table
                       const int*   __restrict__ res_idx,    // dst -> residual row
                       float*       __restrict__ out)        // [n_dst, 128]
{
    __shared__ float ldsS[2][STAGE_K][ROW_F];   // raw src rows   (2 x 8.25 KB)
    __shared__ float ldsR[2][STAGE_K][ROW_F];   // raw rel rows   (2 x 8.25 KB)
    __shared__ float ldsN[2][STAGE_K];          // per-edge norms
    __shared__ float ldsRes[TILE_M][ROW_F];     // residual in / D-tile out

    const int lane  = threadIdx.x;              // 0..31 (EXEC all ones for WMMA)
    const int dst0  = blockIdx.x * TILE_M;
    const int e0    = dst0 * FAN;
    const int m     = lane & 15;
    const int kh    = lane >> 4;

    const unsigned ldsS_off = (unsigned)(uintptr_t)&ldsS[0][0][0];
    const unsigned ldsR_off = (unsigned)(uintptr_t)&ldsR[0][0][0];
    const unsigned voff     = (unsigned)lane * 16u;
    const float inv_in_sqrt = 0.3162277660168379f;   // 1/sqrt(FAN), in-deg == FAN

    // ---- issue one stage of async row gathers ------------------------------
    // Index fetch is vectorized: lane handles edge (lane&15) of the stage
    // (all 32 lanes active, lanes 16..31 duplicate 0..15 -> no EXEC churn),
    // then row ids are broadcast to SGPRs via v_readlane for GVS addressing.
    auto stage_issue = [&](int st) {
        const int sb = st & 1;
        const int i  = lane & 15;
        int e   = e0 + st * STAGE_K + i;
        int sid = src_perm[e];
        int row = HAS_MAP ? src_map[sid] : sid;
        int dg  = deg[sid];
        ldsN[sb][i] = rsqrtf((float)(dg < 1 ? 1 : dg)) * inv_in_sqrt;
        int rrow = HAS_REL ? rel_idx[e] : 0;

        const unsigned lbS = ldsS_off + (unsigned)(sb * STAGE_K) * ROW_BYTES + voff;
        const unsigned lbR = ldsR_off + (unsigned)(sb * STAGE_K) * ROW_BYTES + voff;
#pragma unroll
        for (int ii = 0; ii < STAGE_K; ++ii) {
            int r = __builtin_amdgcn_readlane(row, ii);     // SGPR row id
            unsigned long long gs =
                (unsigned long long)(uintptr_t)(src_rows + (size_t)r * D_DIM);
            async_row_b128(lbS + (unsigned)ii * ROW_BYTES, voff, gs);
            if (HAS_REL) {
                int rr = __builtin_amdgcn_readlane(rrow, ii);
                unsigned long long gr =
                    (unsigned long long)(uintptr_t)(rel_table + (size_t)rr * D_DIM);
                async_row_b128(lbR + (unsigned)ii * ROW_BYTES, voff, gr);
            }
        }
    };

    // Kick off stage 0 immediately so the gathers fly during residual setup.
    stage_issue(0);

    // ---- stage residual rows (coalesced float4) and seed C -----------------
    if (HAS_RES) {
        for (int i = 0; i < TILE_M; ++i) {
            int r = res_idx[dst0 + i];
            float4 v = *(const float4*)(res_rows + (size_t)r * D_DIM + lane * 4);
            *(float4*)(&ldsRes[i][lane * 4]) = v;
        }
    } else {
        float4 z = make_float4(0.f, 0.f, 0.f, 0.f);
        for (int i = 0; i < TILE_M; ++i)
            *(float4*)(&ldsRes[i][lane * 4]) = z;
    }
    __syncthreads();

    v8f c[8];
#pragma unroll
    for (int t = 0; t < 8; ++t) {
#pragma unroll
        for (int j = 0; j < 8; ++j)
            c[t][j] = ldsRes[j + 8 * kh][t * 16 + m];
    }

    // ---- double-buffered pipeline: async gather (s+1) overlaps WMMA (s) ----
    for (int s = 0; s < NSTAGES; ++s) {
        const int sb = s & 1;
        if (s + 1 < NSTAGES) {
            // buffer (s+1)&1 was last read two stages ago; make sure those
            // LDS reads retired before the async engine overwrites it.
            asm volatile("s_wait_dscnt 0x0" ::: "memory");
            stage_issue(s + 1);
            // async loads complete in order: cnt<=N_next => stage s landed.
            if (HAS_REL) asm volatile("s_wait_asynccnt 0x20" ::: "memory");
            else         asm volatile("s_wait_asynccnt 0x10" ::: "memory");
        } else {
            asm volatile("s_wait_asynccnt 0x0" ::: "memory");
        }

        // ---- 4 WMMA chunks of K=4 over the 16 staged edges -----------------
#pragma unroll
        for (int q = 0; q < 4; ++q) {
            const int kl = q * 4;                 // stage-local base row
            const int kb = s * STAGE_K + kl;      // tile-edge index
            const int k0 = kb + 2 * kh;
            const int r0 = kl + 2 * kh;
            v2f a;                                 // selection * per-edge norm
            a.x = ((k0     / FAN) == m) ? ldsN[sb][r0]     : 0.0f;
            a.y = (((k0+1) / FAN) == m) ? ldsN[sb][r0 + 1] : 0.0f;
#pragma unroll
            for (int t = 0; t < 8; ++t) {
                const int col = t * 16 + m;
                float s0 = ldsS[sb][r0][col];
                float s1 = ldsS[sb][r0 + 1][col];
                v2f b;
                if (HAS_REL) {
                    b.x = s0 * ldsR[sb][r0][col];
                    b.y = s1 * ldsR[sb][r0 + 1][col];
                } else {
                    b.x = s0;
                    b.y = s1;
                }
                c[t] = __builtin_amdgcn_wmma_f32_16x16x4_f32(
                    /*neg_a=*/false, a, /*neg_b=*/false, b,
                    /*c_mod=*/(short)0, c[t],
                    /*reuse_a=*/false, /*reuse_b=*/false);
            }
        }
    }

    // ---- transpose D through LDS for coalesced float4 stores ---------------
    __syncthreads();
#pragma unroll
    for (int t = 0; t < 8; ++t) {
#pragma unroll
        for (int j = 0; j < 8; ++j)
            ldsRes[j + 8 * kh][t * 16 + m] = c[t][j];
    }
    __syncthreads();
    for (int i = 0; i < TILE_M; ++i) {
        float4 v = *(const float4*)(&ldsRes[i][lane * 4]);
        *(float4*)(out + (size_t)(dst0 + i) * D_DIM + lane * 4) = v;
    }
}

// ---------------------------------------------------------------------------
// n = g2[global_seed_idx] * LAM + h2 * (1 - LAM)     (LAM = 0.5)
// ---------------------------------------------------------------------------
__global__ __launch_bounds__(32)
void combine_n_kernel(const float* __restrict__ g2, const float* __restrict__ h2,
                      const int* __restrict__ seed, float* __restrict__ out) {
    int d = blockIdx.x, l = threadIdx.x;
    int s = seed[d];
    float4 a = *(const float4*)(g2 + (size_t)s * D_DIM + l * 4);
    float4 b = *(const float4*)(h2 + (size_t)d * D_DIM + l * 4);
    float4 o;
    o.x = a.x * 0.5f + b.x * 0.5f;  o.y = a.y * 0.5f + b.y * 0.5f;
    o.z = a.z * 0.5f + b.z * 0.5f;  o.w = a.w * 0.5f + b.w * 0.5f;
    *(float4*)(out + (size_t)d * D_DIM + l * 4) = o;
}

// ---------------------------------------------------------------------------
// e_out[d] = r1[s]*LAM2 + edge_table[rel_idx_l1[s]]*(1-LAM2), s = rel_seed[d]
// ---------------------------------------------------------------------------
__global__ __launch_bounds__(32)
void combine_e_kernel(const float* __restrict__ r1, const float* __restrict__ edge_table,
                      const int* __restrict__ rel_idx_l1, const int* __restrict__ rel_seed,
                      float* __restrict__ out) {
    int d = blockIdx.x, l = threadIdx.x;
    int s  = rel_seed[d];
    int er = rel_idx_l1[s];
    float4 a = *(const float4*)(r1 + (size_t)s * D_DIM + l * 4);
    float4 b = *(const float4*)(edge_table + (size_t)er * D_DIM + l * 4);
    float4 o;
    o.x = a.x * 0.5f + b.x * 0.5f;  o.y = a.y * 0.5f + b.y * 0.5f;
    o.z = a.z * 0.5f + b.z * 0.5f;  o.w = a.w * 0.5f + b.w * 0.5f;
    *(float4*)(out + (size_t)d * D_DIM + l * 4) = o;
}

// ---------------------------------------------------------------------------
// kernel_launch
// ---------------------------------------------------------------------------
extern "C" void kernel_launch(void* const* d_in, const int* in_sizes, int n_in,
                              void* d_out, int out_size, void* d_ws, size_t ws_size,
                              hipStream_t stream) {
    (void)in_sizes; (void)n_in; (void)out_size; (void)ws_size;

    const float* global_table = (const float*)d_in[0];
    const float* node_table   = (const float*)d_in[1];
    const float* edge_table   = (const float*)d_in[2];
    const int* node_idx_l0    = (const int*)d_in[3];
    const int* node_idx_l1    = (const int*)d_in[4];
    const int* node_idx_l2    = (const int*)d_in[5];
    const int* b0_src         = (const int*)d_in[6];
    const int* b0_rel         = (const int*)d_in[8];
    const int* b1_src         = (const int*)d_in[9];
    const int* b1_rel         = (const int*)d_in[11];
    const int* gnode_idx_l0   = (const int*)d_in[12];
    const int* gnode_idx_l1   = (const int*)d_in[13];
    const int* gnode_idx_l2   = (const int*)d_in[14];
    const int* gb0_src        = (const int*)d_in[15];
    const int* gb0_rel        = (const int*)d_in[17];
    const int* gb1_src        = (const int*)d_in[18];
    const int* gb1_rel        = (const int*)d_in[20];
    const int* global_seed    = (const int*)d_in[21];
    const int* rel_idx_l0     = (const int*)d_in[22];
    const int* rel_idx_l1     = (const int*)d_in[23];
    const int* rb_src         = (const int*)d_in[24];
    const int* rel_seed       = (const int*)d_in[26];

    // workspace layout (~12.2 MB): h1, g1, h2, g2, r1, then int degree arrays
    float* h1 = (float*)d_ws;                        // N1*D
    float* g1 = h1 + (size_t)N_N1 * D_DIM;           // N1*D
    float* h2 = g1 + (size_t)N_N1 * D_DIM;           // B*D
    float* g2 = h2 + (size_t)N_B  * D_DIM;           // B*D
    float* r1 = g2 + (size_t)N_B  * D_DIM;           // RS*D
    int* deg_b0  = (int*)(r1 + (size_t)N_RS * D_DIM);
    int* deg_b1  = deg_b0  + N_N0;
    int* deg_gb0 = deg_b1  + N_N1;
    int* deg_gb1 = deg_gb0 + N_N0;
    int* deg_rb  = deg_gb1 + N_N1;
    const size_t deg_total = 2 * (size_t)(N_N0 + N_N1) + N_R0;

    hipMemsetAsync(deg_b0, 0, deg_total * sizeof(int), stream);

    const int CB = 256;
    count_deg_kernel<<<(N_N0 + CB - 1) / CB, CB, 0, stream>>>(b0_src,  deg_b0,  N_N0);
    count_deg_kernel<<<(N_N1 + CB - 1) / CB, CB, 0, stream>>>(b1_src,  deg_b1,  N_N1);
    count_deg_kernel<<<(N_N0 + CB - 1) / CB, CB, 0, stream>>>(gb0_src, deg_gb0, N_N0);
    count_deg_kernel<<<(N_N1 + CB - 1) / CB, CB, 0, stream>>>(gb1_src, deg_gb1, N_N1);
    count_deg_kernel<<<(N_R0 + CB - 1) / CB, CB, 0, stream>>>(rb_src,  deg_rb,  N_R0);

    // temporal flow: h1 = block(node[idx0], edge[b0_rel]) + node[idx1]
    block_wmma_kernel<true, true, true><<<N_N1 / TILE_M, 32, 0, stream>>>(
        node_table, node_idx_l0, b0_src, edge_table, b0_rel, deg_b0,
        node_table, node_idx_l1, h1);
    // h2 = block(h1, edge[b1_rel]) + node[idx2]
    block_wmma_kernel<false, true, true><<<N_B / TILE_M, 32, 0, stream>>>(
        h1, nullptr, b1_src, edge_table, b1_rel, deg_b1,
        node_table, node_idx_l2, h2);
    // global flow
    block_wmma_kernel<true, true, true><<<N_N1 / TILE_M, 32, 0, stream>>>(
        global_table, gnode_idx_l0, gb0_src, edge_table, gb0_rel, deg_gb0,
        global_table, gnode_idx_l1, g1);
    block_wmma_kernel<false, true, true><<<N_B / TILE_M, 32, 0, stream>>>(
        g1, nullptr, gb1_src, edge_table, gb1_rel, deg_gb1,
        global_table, gnode_idx_l2, g2);
    // relation flow: r1 = block(edge[rel_idx_l0], None) — no gating, no residual
    block_wmma_kernel<true, false, false><<<N_RS / TILE_M, 32, 0, stream>>>(
        edge_table, rel_idx_l0, rb_src, nullptr, nullptr, deg_rb,
        nullptr, nullptr, r1);

    float* out_n = (float*)d_out;
    float* out_e = out_n + (size_t)N_B * D_DIM;
    combine_n_kernel<<<N_B, 32, 0, stream>>>(g2, h2, global_seed, out_n);
    combine_e_kernel<<<N_B, 32, 0, stream>>>(r1, edge_table, rel_idx_l1, rel_seed, out_e);
}